// LatentODE_69965017252165
// MI455X (gfx1250) — compile-verified
//
#include <hip/hip_runtime.h>
#include <math.h>

typedef float v2f __attribute__((ext_vector_type(2)));
typedef float v8f __attribute__((ext_vector_type(8)));

#define B_SZ 8192
#define T_SZ 200
#define D_SZ 3
#define H_SZ 20
#define L_SZ 10

#define WAVES_PER_BLOCK 4

// D = A(16x4) * B(4x16) + C(16x16), all fp32. wave32.
__device__ __forceinline__ v8f wmma4(v2f a, v2f b, v8f c) {
  return __builtin_amdgcn_wmma_f32_16x16x4_f32(false, a, false, b, (short)0, c,
                                               false, false);
}

// Hardware transcendental tanh (V_TANH_F32 trans op on CDNA5).
__device__ __forceinline__ float fast_tanh(float x) {
#if __has_builtin(__builtin_amdgcn_tanhf)
  return __builtin_amdgcn_tanhf(x);
#else
  float r;
  asm volatile("v_tanh_f32 %0, %1\n\tv_nop" : "=v"(r) : "v"(x));
  return r;
#endif
}

// Half-wave exchange: lane <-> lane^16 via ds_swizzle SWAPX16 (no LDS memory).
__device__ __forceinline__ float swap16(float x) {
  return __builtin_bit_cast(
      float, __builtin_amdgcn_ds_swizzle(__builtin_bit_cast(int, x), 0x401F));
}

__global__ __launch_bounds__(WAVES_PER_BLOCK * 32)
void latent_ode_kernel(const float* __restrict__ ts, const float* __restrict__ ys,
                       const float* __restrict__ eps,
                       const float* __restrict__ encW1, const float* __restrict__ encB1,
                       const float* __restrict__ encW2, const float* __restrict__ encB2,
                       const float* __restrict__ encW3, const float* __restrict__ encB3,
                       const float* __restrict__ l2hW, const float* __restrict__ l2hB,
                       const float* __restrict__ h2dW, const float* __restrict__ h2dB,
                       const float* __restrict__ fW1, const float* __restrict__ fB1,
                       const float* __restrict__ fW2, const float* __restrict__ fB2,
                       float* __restrict__ pred, float* __restrict__ outMean,
                       float* __restrict__ outStd)
{
  const int tid  = (int)threadIdx.x;
  const int wave = tid >> 5;
  const int lane = tid & 31;
  const int n16  = lane & 15;
  const int hi   = lane >> 4;              // half-wave: 0 or 1
  const bool h   = (hi != 0);
  const int rowBase = ((int)blockIdx.x * WAVES_PER_BLOCK + wave) * 16;
  const int row = rowBase + n16;           // lanes 16..31 duplicate rows 0..15

  // ---------------- encoder (scalar per-lane; both halves duplicate) --------
  float yin[D_SZ];
#pragma unroll
  for (int d = 0; d < D_SZ; ++d)
    yin[d] = ys[(size_t)row * T_SZ * D_SZ + d];

  float h1[H_SZ];
#pragma unroll
  for (int j = 0; j < H_SZ; ++j) {
    float acc = encB1[j];
#pragma unroll
    for (int d = 0; d < D_SZ; ++d) acc += yin[d] * encW1[d * H_SZ + j];
    h1[j] = fmaxf(acc, 0.0f);
  }
  float h2[H_SZ];
#pragma unroll
  for (int j = 0; j < H_SZ; ++j) {
    float acc = encB2[j];
#pragma unroll
    for (int i = 0; i < H_SZ; ++i) acc += h1[i] * encW2[i * H_SZ + j];
    h2[j] = fmaxf(acc, 0.0f);
  }
  float latent[L_SZ];
#pragma unroll
  for (int j = 0; j < L_SZ; ++j) {
    float m  = encB3[j];
    float ls = encB3[L_SZ + j];
#pragma unroll
    for (int i = 0; i < H_SZ; ++i) {
      m  += h2[i] * encW3[i * (2 * L_SZ) + j];
      ls += h2[i] * encW3[i * (2 * L_SZ) + L_SZ + j];
    }
    const float sd = expf(ls);
    latent[j] = m + eps[(size_t)row * L_SZ + j] * sd;
    if (lane < 16) {
      outMean[(size_t)row * L_SZ + j] = m;
      outStd [(size_t)row * L_SZ + j] = sd;
    }
  }
  float y0v[H_SZ];
#pragma unroll
  for (int j = 0; j < H_SZ; ++j) {
    float acc = l2hB[j];
#pragma unroll
    for (int i = 0; i < L_SZ; ++i) acc += latent[i] * l2hW[i * H_SZ + j];
    y0v[j] = fmaxf(acc, 0.0f);
  }

  // ---- transposed state Z = y^T in C-fragment layout ----
  // zc0[v]: feature v+8*hi, batch n16;  zc1[v]: feature 16+v+8*hi (valid hi=0,v<4)
  v8f yc0, yc1;
#pragma unroll
  for (int v = 0; v < 8; ++v) {
    yc0[v] = h ? y0v[v + 8] : y0v[v];
    yc1[v] = (!h && v < 4) ? y0v[16 + v] : 0.0f;
  }

  // ---- weights as A-fragments; index 5 = bias column at K=20 (hi=0 only) ----
  v2f w1f[2][6], w2f[2][6], w3f[6];
#pragma unroll
  for (int s = 0; s < 5; ++s) {
    const int k0 = 4 * s + 2 * hi;
#pragma unroll
    for (int nt = 0; nt < 2; ++nt) {
      const int n = 16 * nt + n16;                 // output feature (M)
      const bool ok = n < H_SZ;
      w1f[nt][s].x = ok ? fW1[k0 * H_SZ + n]       : 0.0f;
      w1f[nt][s].y = ok ? fW1[(k0 + 1) * H_SZ + n] : 0.0f;
      w2f[nt][s].x = ok ? fW2[k0 * H_SZ + n]       : 0.0f;
      w2f[nt][s].y = ok ? fW2[(k0 + 1) * H_SZ + n] : 0.0f;
    }
    const bool ok3 = n16 < D_SZ;
    w3f[s].x = ok3 ? h2dW[k0 * D_SZ + n16]       : 0.0f;
    w3f[s].y = ok3 ? h2dW[(k0 + 1) * D_SZ + n16] : 0.0f;
  }
#pragma unroll
  for (int nt = 0; nt < 2; ++nt) {
    const int n = 16 * nt + n16;
    const bool ok = (!h) && (n < H_SZ);
    w1f[nt][5].x = ok ? fB1[n] : 0.0f;  w1f[nt][5].y = 0.0f;
    w2f[nt][5].x = ok ? fB2[n] : 0.0f;  w2f[nt][5].y = 0.0f;
  }
  w3f[5].x = (!h && n16 < D_SZ) ? h2dB[n16] : 0.0f;
  w3f[5].y = 0.0f;
  // B-operand "ones row" at K=20 (shared by all bias steps, data-independent)
  v2f bone;
  bone.x = h ? 0.0f : 1.0f;
  bone.y = 0.0f;
  const v8f vzero = {};

  // ---------------- helpers ----------------
  // Build B-operand k-slices (K=feature, N=batch) from a C-layout Z:
  // pure half-wave swizzles + selects, no LDS memory.
  auto buildB = [&](const v8f& z0, const v8f& z1, v2f* bf) {
    const float s0 = swap16(z0[0]), s1 = swap16(z0[1]);
    const float s2 = swap16(z0[2]), s3 = swap16(z0[3]);
    const float s4 = swap16(z0[4]), s5 = swap16(z0[5]);
    const float s6 = swap16(z0[6]), s7 = swap16(z0[7]);
    const float t2 = swap16(z1[2]), t3 = swap16(z1[3]);
    bf[0].x = h ? s2    : z0[0];  bf[0].y = h ? s3    : z0[1];  // K 0,1 / 2,3
    bf[1].x = h ? s6    : z0[4];  bf[1].y = h ? s7    : z0[5];  // K 4,5 / 6,7
    bf[2].x = h ? z0[2] : s0;     bf[2].y = h ? z0[3] : s1;     // K 8,9 / 10,11
    bf[3].x = h ? z0[6] : s4;     bf[3].y = h ? z0[7] : s5;     // K 12,13 / 14,15
    bf[4].x = h ? t2    : z1[0];  bf[4].y = h ? t3    : z1[1];  // K 16,17 / 18,19
  };
  // 6-deep chain: bias step first (C = 0), then 5 data steps.
  auto matmul = [&](const v2f (*wa)[6], const v2f* bf, v8f& o0, v8f& o1) {
    o0 = wmma4(wa[0][5], bone, vzero);
    o1 = wmma4(wa[1][5], bone, vzero);
#pragma unroll
    for (int s = 0; s < 5; ++s) {
      o0 = wmma4(wa[0][s], bf[s], o0);
      o1 = wmma4(wa[1][s], bf[s], o1);
    }
  };
  auto fEval = [&](const v2f* bf, v8f& k0, v8f& k1) {
    v8f t0, t1;
    matmul(w1f, bf, t0, t1);
#pragma unroll
    for (int v = 0; v < 8; ++v) {
      t0[v] = fast_tanh(t0[v]);
      t1[v] = fast_tanh(t1[v]);
    }
    v2f bt[5];
    buildB(t0, t1, bt);
    matmul(w2f, bt, k0, k1);
  };
  auto project = [&](const v2f* bf, int t) {
    v8f p = wmma4(w3f[5], bone, vzero);
#pragma unroll
    for (int s = 0; s < 5; ++s) p = wmma4(w3f[s], bf[s], p);
    if (!h) {
      // pred^T rows d=0..2 live in VGPRs 0..2 of the hi=0 half:
      // 3 contiguous floats per batch row.
      float* o = &pred[((size_t)(rowBase + n16) * T_SZ + t) * D_SZ];
      o[0] = p[0]; o[1] = p[1]; o[2] = p[2];
    }
  };

  // ---------------- RK4 time loop (state resident in fragments) -------------
  // Incremental accumulation keeps only {y, acc, k, u} live (not k1..k4).
  for (int t = 0; t < T_SZ - 1; ++t) {
    const float dt  = ts[t + 1] - ts[t];
    const float hdt = 0.5f * dt;
    v2f bf[5];

    buildB(yc0, yc1, bf);
    project(bf, t);                        // pred_ys at time t (reuses y's B frags)

    v8f ka, kb, acc0, acc1, u0, u1;
    fEval(bf, ka, kb);                     // k1
#pragma unroll
    for (int v = 0; v < 8; ++v) {
      acc0[v] = ka[v];                 acc1[v] = kb[v];
      u0[v] = yc0[v] + hdt * ka[v];    u1[v] = yc1[v] + hdt * kb[v];
    }
    buildB(u0, u1, bf);
    fEval(bf, ka, kb);                     // k2
#pragma unroll
    for (int v = 0; v < 8; ++v) {
      acc0[v] += 2.0f * ka[v];         acc1[v] += 2.0f * kb[v];
      u0[v] = yc0[v] + hdt * ka[v];    u1[v] = yc1[v] + hdt * kb[v];
    }
    buildB(u0, u1, bf);
    fEval(bf, ka, kb);                     // k3
#pragma unroll
    for (int v = 0; v < 8; ++v) {
      acc0[v] += 2.0f * ka[v];         acc1[v] += 2.0f * kb[v];
      u0[v] = yc0[v] + dt * ka[v];     u1[v] = yc1[v] + dt * kb[v];
    }
    buildB(u0, u1, bf);
    fEval(bf, ka, kb);                     // k4

    const float sdt = dt * (1.0f / 6.0f);
#pragma unroll
    for (int v = 0; v < 8; ++v) {
      yc0[v] += sdt * (acc0[v] + ka[v]);
      yc1[v] += sdt * (acc1[v] + kb[v]);
    }
  }
  // final timestep projection
  {
    v2f bf[5];
    buildB(yc0, yc1, bf);
    project(bf, T_SZ - 1);
  }
}

extern "C" void kernel_launch(void* const* d_in, const int* in_sizes, int n_in,
                              void* d_out, int out_size, void* d_ws, size_t ws_size,
                              hipStream_t stream) {
  const float* ts    = (const float*)d_in[0];
  const float* ys    = (const float*)d_in[1];
  const float* eps   = (const float*)d_in[2];
  const float* encW1 = (const float*)d_in[3];
  const float* encB1 = (const float*)d_in[4];
  const float* encW2 = (const float*)d_in[5];
  const float* encB2 = (const float*)d_in[6];
  const float* encW3 = (const float*)d_in[7];
  const float* encB3 = (const float*)d_in[8];
  const float* l2hW  = (const float*)d_in[9];
  const float* l2hB  = (const float*)d_in[10];
  const float* h2dW  = (const float*)d_in[11];
  const float* h2dB  = (const float*)d_in[12];
  const float* fW1   = (const float*)d_in[13];
  const float* fB1   = (const float*)d_in[14];
  const float* fW2   = (const float*)d_in[15];
  const float* fB2   = (const float*)d_in[16];

  float* pred = (float*)d_out;
  float* mean = pred + (size_t)B_SZ * T_SZ * D_SZ;
  float* stdv = mean + (size_t)B_SZ * L_SZ;

  dim3 grid(B_SZ / (WAVES_PER_BLOCK * 16));   // 128 blocks = 512 waves
  dim3 block(WAVES_PER_BLOCK * 32);           // 128 threads
  hipLaunchKernelGGL(latent_ode_kernel, grid, block, 0, stream,
                     ts, ys, eps, encW1, encB1, encW2, encB2, encW3, encB3,
                     l2hW, l2hB, h2dW, h2dB, fW1, fB1, fW2, fB2,
                     pred, mean, stdv);
}